// MSBEGCL_Encoder_27994596835373
// MI455X (gfx1250) — compile-verified
//
#include <hip/hip_runtime.h>
#include <hip/hip_bf16.h>

#define USER_NUM 100000
#define ITEM_NUM 50000
#define NTOT     (USER_NUM + ITEM_NUM)   /* 150000 */
#define EMB      64
#define NNZ_CNT  3000000
#define N_LAYERS 3
#define EPS_F    0.1f

typedef float v2f __attribute__((ext_vector_type(2)));
typedef float v8f __attribute__((ext_vector_type(8)));

// ---------------------------------------------------------------------------
// Kernel 1: concat [user_emb; item_emb] -> ego, and initialize acc = ego.
// float4 vectorized; NTOT*EMB/4 = 2.4M threads.
// ---------------------------------------------------------------------------
__global__ void k_init(const float* __restrict__ user_emb,
                       const float* __restrict__ item_emb,
                       float* __restrict__ ego,
                       float* __restrict__ acc) {
    int i = blockIdx.x * blockDim.x + threadIdx.x;          // float4 index
    const int total  = NTOT * EMB / 4;
    const int ubound = USER_NUM * EMB / 4;
    if (i >= total) return;
    float4 v = (i < ubound) ? ((const float4*)user_emb)[i]
                            : ((const float4*)item_emb)[i - ubound];
    ((float4*)ego)[i] = v;
    ((float4*)acc)[i] = v;
}

// ---------------------------------------------------------------------------
// Kernel 2: per-row inverse L2 norms of noise via V_WMMA_F32_16X16X4_F32.
// One wave32 handles 16 rows: diag(X * X^T) over K=64 with 16 accumulating
// WMMAs. For X*X^T the A and B VGPR layouts coincide, so one v2f feeds both.
//   A/B layout: lane<16 -> (M=lane, K=kk+0 / kk+1); lane>=16 -> (M=lane-16,
//               K=kk+2 / kk+3).
//   D diag:     m<8 -> c[m] @ lane m ; m>=8 -> c[m-8] @ lane m+16.
// Writes inv_scale[row] = EPS / max(sqrt(diag), 1e-12).
// ---------------------------------------------------------------------------
__global__ void k_noise_invnorm(const float* __restrict__ noise,
                                float* __restrict__ inv_scale) {
    const int ntiles = (N_LAYERS * NTOT) / 16;              // 28125
    int wave = (blockIdx.x * blockDim.x + threadIdx.x) >> 5; // wave-uniform
    int lane = threadIdx.x & 31;
    if (wave >= ntiles) return;                             // uniform per wave

    int  base = wave * 16;
    int  row  = base + (lane & 15);
    int  kb   = (lane >> 4) * 2;                            // 0 or 2
    const float* p = noise + (size_t)row * EMB + kb;

    v8f c = {};
    #pragma unroll
    for (int kk = 0; kk < EMB; kk += 4) {
        v2f a;
        a.x = p[kk + 0];
        a.y = p[kk + 1];
        // D = A * A^T + C   (exact f32 FMA path on the matrix pipe)
        c = __builtin_amdgcn_wmma_f32_16x16x4_f32(
                /*neg_a=*/false, a, /*neg_b=*/false, a,
                /*c_mod=*/(short)0, c, /*reuse_a=*/false, /*reuse_b=*/false);
    }

    // Extract the diagonal element owned by this lane (if any).
    int sel = (lane < 8) ? lane : ((lane >= 24) ? (lane - 24) : -1);
    float d = 0.0f;
    #pragma unroll
    for (int j = 0; j < 8; ++j) d = (sel == j) ? c[j] : d;

    if (sel >= 0) {
        int m = (lane < 8) ? lane : (lane - 16);            // 0..15
        float n = sqrtf(d);
        inv_scale[base + m] = EPS_F / fmaxf(n, 1e-12f);
    }
}

// ---------------------------------------------------------------------------
// Kernel 3: COO SpMM scatter, Y += vals[e] * X[cols[e], :] into row rows[e].
// One wave32 per edge-group: each lane owns a float2 (64 floats / 32 lanes),
// scatter via global_atomic_add_f32 (L2-resident output).
// ---------------------------------------------------------------------------
#define EDGES_PER_WAVE 4
__global__ void k_spmm(const float* __restrict__ vals,
                       const int*   __restrict__ rows,
                       const int*   __restrict__ cols,
                       const float* __restrict__ x,
                       float* __restrict__ y) {
    int wave = (blockIdx.x * blockDim.x + threadIdx.x) >> 5;
    int lane = threadIdx.x & 31;
    long e0 = (long)wave * EDGES_PER_WAVE;
    if (e0 >= NNZ_CNT) return;

    // Stream-prefetch the upcoming edge index/value data (speculative).
    __builtin_prefetch(cols + e0 + 2048, 0, 1);
    __builtin_prefetch(rows + e0 + 2048, 0, 1);
    __builtin_prefetch(vals + e0 + 2048, 0, 1);

    #pragma unroll
    for (int j = 0; j < EDGES_PER_WAVE; ++j) {
        long e = e0 + j;
        if (e >= NNZ_CNT) return;
        int   r = rows[e];
        int   c = cols[e];
        float v = vals[e];
        float2 xv = ((const float2*)(x + (size_t)c * EMB))[lane];
        float* dst = y + (size_t)r * EMB + lane * 2;
        unsafeAtomicAdd(dst + 0, v * xv.x);   // global_atomic_add_f32
        unsafeAtomicAdd(dst + 1, v * xv.y);
    }
}

// ---------------------------------------------------------------------------
// Kernel 4: ego = Y + sign(Y) * noise_k * inv_scale_k[row];
//           acc += ego; optionally acc *= 0.25 on the last layer.
// ---------------------------------------------------------------------------
__device__ __forceinline__ float signf_(float v) {
    return (v > 0.0f) ? 1.0f : ((v < 0.0f) ? -1.0f : 0.0f);
}

__global__ void k_fuse(const float* __restrict__ y,
                       const float* __restrict__ noise_k,
                       const float* __restrict__ inv_scale_k,
                       float* __restrict__ ego,
                       float* __restrict__ acc,
                       int final_scale) {
    int i = blockIdx.x * blockDim.x + threadIdx.x;          // float4 index
    const int total = NTOT * EMB / 4;
    if (i >= total) return;
    int row = i >> 4;                                       // 16 float4 / row
    float is = inv_scale_k[row];

    float4 yv = ((const float4*)y)[i];
    float4 nv = ((const float4*)noise_k)[i];

    float4 ev;
    ev.x = yv.x + signf_(yv.x) * nv.x * is;
    ev.y = yv.y + signf_(yv.y) * nv.y * is;
    ev.z = yv.z + signf_(yv.z) * nv.z * is;
    ev.w = yv.w + signf_(yv.w) * nv.w * is;
    ((float4*)ego)[i] = ev;

    float4 av = ((float4*)acc)[i];
    av.x += ev.x; av.y += ev.y; av.z += ev.z; av.w += ev.w;
    if (final_scale) {
        av.x *= 0.25f; av.y *= 0.25f; av.z *= 0.25f; av.w *= 0.25f;
    }
    ((float4*)acc)[i] = av;
}

// ---------------------------------------------------------------------------
extern "C" void kernel_launch(void* const* d_in, const int* in_sizes, int n_in,
                              void* d_out, int out_size, void* d_ws, size_t ws_size,
                              hipStream_t stream) {
    const float* user_emb = (const float*)d_in[0];
    const float* item_emb = (const float*)d_in[1];
    const float* adj_vals = (const float*)d_in[2];
    const float* noise    = (const float*)d_in[3];  // [3, N, 64]
    const int*   adj_rows = (const int*)d_in[4];
    const int*   adj_cols = (const int*)d_in[5];

    float* acc = (float*)d_out;                     // [N, 64] accumulator/out

    float* ego = (float*)d_ws;                      // [N, 64]
    float* Y   = ego + (size_t)NTOT * EMB;          // [N, 64]
    float* inv = Y   + (size_t)NTOT * EMB;          // [3, N]

    const int elem4 = NTOT * EMB / 4;               // 2.4M float4s

    // 1) ego = concat(user, item); acc = ego
    k_init<<<(elem4 + 255) / 256, 256, 0, stream>>>(user_emb, item_emb, ego, acc);

    // 2) all-layer noise inverse norms via WMMA
    {
        int waves  = (N_LAYERS * NTOT) / 16;        // 28125
        int blocks = (waves + 7) / 8;               // 8 waves / block
        k_noise_invnorm<<<blocks, 256, 0, stream>>>(noise, inv);
    }

    // 3) three propagation layers
    for (int k = 0; k < N_LAYERS; ++k) {
        hipMemsetAsync(Y, 0, (size_t)NTOT * EMB * sizeof(float), stream);

        long waves  = (NNZ_CNT + EDGES_PER_WAVE - 1) / EDGES_PER_WAVE; // 750000
        int  blocks = (int)((waves + 7) / 8);                          // 93750
        k_spmm<<<blocks, 256, 0, stream>>>(adj_vals, adj_rows, adj_cols, ego, Y);

        k_fuse<<<(elem4 + 255) / 256, 256, 0, stream>>>(
            Y,
            noise + (size_t)k * NTOT * EMB,
            inv   + (size_t)k * NTOT,
            ego, acc, (k == N_LAYERS - 1) ? 1 : 0);
    }
}